// MaxAttention_7928509629191
// MI455X (gfx1250) — compile-verified
//
#include <hip/hip_runtime.h>
#include <hip/hip_bf16.h>

#define BB      64
#define SS      4096
#define DD      256
#define LABELS  50

typedef __attribute__((ext_vector_type(2))) float v2f;
typedef __attribute__((ext_vector_type(4))) float v4f;
typedef __attribute__((ext_vector_type(8))) float v8f;

__device__ __forceinline__ v4f vmax4(v4f a, v4f b) {
    v4f r;
    r.x = fmaxf(a.x, b.x);
    r.y = fmaxf(a.y, b.y);
    r.z = fmaxf(a.z, b.z);
    r.w = fmaxf(a.w, b.w);
    return r;
}

// ---------------------------------------------------------------------------
// Kernel A: streaming partial segment-max.
// grid = (split_cnt, B), block = 64 threads. Thread t owns float4 column t
// (d = 4t..4t+3) and streams rows [split*chunk, min((split+1)*chunk, len)).
// Non-temporal b128 loads: data is read exactly once and exceeds L2 (268MB).
// ---------------------------------------------------------------------------
__global__ void seg_partial_max(const float* __restrict__ code,
                                const int*   __restrict__ lengths,
                                float*       __restrict__ partial,
                                int split_cnt, int chunk) {
    const int split = blockIdx.x;
    const int b     = blockIdx.y;
    const int d4    = threadIdx.x;            // 0..63

    const int len = lengths[b];
    const int s0  = split * chunk;
    int s1        = s0 + chunk;
    if (s1 > len) s1 = len;

    const float ninf = -__builtin_huge_valf();
    v4f m; m.x = ninf; m.y = ninf; m.z = ninf; m.w = ninf;

    const v4f* base = (const v4f*)(code + (size_t)b * SS * DD) + d4; // row stride = 64 v4f
    for (int s = s0; s < s1; ++s) {
        v4f v = __builtin_nontemporal_load(base + (size_t)s * (DD / 4));
        m = vmax4(m, v);
    }

    v4f* out = (v4f*)(partial + ((size_t)b * split_cnt + split) * DD) + d4;
    *out = m;
}

// ---------------------------------------------------------------------------
// Kernel B: reduce split partials -> pool[B, D].
// grid = B, block = 256 (thread = one d column).
// ---------------------------------------------------------------------------
__global__ void seg_reduce(const float* __restrict__ partial,
                           float*       __restrict__ pool,
                           int split_cnt) {
    const int b = blockIdx.x;
    const int d = threadIdx.x;
    float m = -__builtin_huge_valf();
    for (int sp = 0; sp < split_cnt; ++sp)
        m = fmaxf(m, partial[((size_t)b * split_cnt + sp) * DD + d]);
    pool[b * DD + d] = m;
}

// ---------------------------------------------------------------------------
// Kernel C: out[64,50] = pool[64,256] @ W_pool[50,256]^T via
// V_WMMA_F32_16X16X4_F32 (full f32 precision, matches reference).
// One block, 16 waves; wave w = (m_tile, n_tile) owns a 16x16 output tile.
// K = 256 -> 64 wmma steps.
//
// Fragment layouts per CDNA5 ISA (7.12.2):
//   A (16x4, 2 VGPRs): lanes 0-15 hold M=lane, v0=K0,v1=K1;
//                      lanes 16-31 hold M=lane-16, v0=K2,v1=K3.
//   B (4x16) mirrored: lanes index N, lane-half selects K pair.
//   C/D (8 VGPRs): VGPR r: lanes 0-15 -> M=r, N=lane; lanes 16-31 -> M=r+8.
// ---------------------------------------------------------------------------
__global__ void pool_gemm_wmma(const float* __restrict__ pool,
                               const float* __restrict__ Wp,
                               float*       __restrict__ out) {
    const int wave = threadIdx.x >> 5;    // 0..15
    const int lane = threadIdx.x & 31;
    const int mt = wave >> 2;
    const int nt = wave & 3;
    const int m_base = mt * 16;
    const int n_base = nt * 16;

    const int half = lane >> 4;           // 0 or 1 -> K pair select
    const int l    = lane & 15;
    const int row  = m_base + l;          // A: M index
    const int col  = n_base + l;          // B/D: N index
    const int colc = (col < LABELS) ? col : (LABELS - 1);  // clamp reads, guard stores

    v8f c = {};
    for (int k0 = 0; k0 < DD; k0 += 4) {
        const int ka = k0 + half * 2;
        v2f a, bfrag;
        a.x     = pool[row * DD + ka];
        a.y     = pool[row * DD + ka + 1];
        bfrag.x = Wp[colc * DD + ka];
        bfrag.y = Wp[colc * DD + ka + 1];
        // (neg_a, A, neg_b, B, c_mod, C, reuse_a, reuse_b)
        c = __builtin_amdgcn_wmma_f32_16x16x4_f32(false, a, false, bfrag,
                                                  (short)0, c, false, false);
    }

    if (col < LABELS) {
#pragma unroll
        for (int r = 0; r < 8; ++r) {
            const int orow = m_base + r + half * 8;
            out[orow * LABELS + col] = c[r];
        }
    }
}

// ---------------------------------------------------------------------------
// Launcher
// ---------------------------------------------------------------------------
extern "C" void kernel_launch(void* const* d_in, const int* in_sizes, int n_in,
                              void* d_out, int out_size, void* d_ws, size_t ws_size,
                              hipStream_t stream) {
    const float* code    = (const float*)d_in[2];  // [B,S,D]
    const int*   lengths = (const int*)  d_in[3];  // [B]
    const float* Wp      = (const float*)d_in[7];  // [LABELS, D]
    float*       out     = (float*)d_out;          // [B, LABELS]

    // Workspace: [partial: B*split_cnt*D floats][pool: B*D floats]
    const size_t poolBytes  = (size_t)BB * DD * sizeof(float);
    const size_t sliceBytes = (size_t)BB * DD * sizeof(float);
    size_t avail = (ws_size > poolBytes) ? (ws_size - poolBytes) : 0;
    int split_cnt = (int)(avail / sliceBytes);
    if (split_cnt > 64) split_cnt = 64;
    if (split_cnt < 1)  split_cnt = 1;
    const int chunk = (SS + split_cnt - 1) / split_cnt;

    float* partial = (float*)d_ws;
    float* pool    = (float*)((char*)d_ws + (size_t)split_cnt * sliceBytes);

    dim3 gridA(split_cnt, BB);
    seg_partial_max<<<gridA, 64, 0, stream>>>(code, lengths, partial, split_cnt, chunk);
    seg_reduce<<<BB, DD, 0, stream>>>(partial, pool, split_cnt);
    pool_gemm_wmma<<<1, 512, 0, stream>>>(pool, Wp, out);
}